// BB_loss_80298708566608
// MI455X (gfx1250) — compile-verified
//
#include <hip/hip_runtime.h>
#include <math.h>

#define BATCH   2
#define CH      3
#define H       144
#define W       144
#define PS      3
#define NPW     48            // patches per row
#define NP      (NPW*NPW)     // 2304 patches per batch
#define G_REAL  14460
#define G_PAD   14464         // multiple of 128
#define D_PAD   32            // 27 features padded to 32
#define CCH     128           // gf columns staged per LDS chunk
#define NCC     (G_PAD/CCH)   // 113 staged chunks
#define LDS_ROW 36            // padded LDS row stride (16 distinct banks)
#define TILES_PER_BATCH (NP/16)           // 144
#define BLOCKS_PER_BATCH (TILES_PER_BATCH/8) // 18
#define A_CUBIC (-0.75f)

typedef __attribute__((ext_vector_type(2))) float v2f;
typedef __attribute__((ext_vector_type(8))) float v8f;
typedef __attribute__((ext_vector_type(4))) int   v4i;

// ------------------------------------------------ async LDS path detection --
#if defined(__has_builtin)
#if __has_builtin(__builtin_amdgcn_global_load_async_to_lds_b128) && \
    __has_builtin(__builtin_amdgcn_global_load_async_to_lds_b32)
#define USE_ASYNC_LDS 1
#endif
#endif

#if defined(USE_ASYNC_LDS)
typedef __attribute__((address_space(1))) v4i gv4i;
typedef __attribute__((address_space(3))) v4i lv4i;
typedef __attribute__((address_space(1))) int gi32;
typedef __attribute__((address_space(3))) int li32;

__device__ __forceinline__ void cp_async_b128(const float* g, float* l) {
    __builtin_amdgcn_global_load_async_to_lds_b128((gv4i*)g, (lv4i*)l, 0, 0);
}
__device__ __forceinline__ void cp_async_b32(const float* g, float* l) {
    __builtin_amdgcn_global_load_async_to_lds_b32((gi32*)g, (li32*)l, 0, 0);
}
__device__ __forceinline__ void wait_async0() {
#if __has_builtin(__builtin_amdgcn_s_wait_asynccnt)
    __builtin_amdgcn_s_wait_asynccnt(0);
#else
    asm volatile("s_wait_asynccnt 0x0" ::: "memory");
#endif
}
#else
__device__ __forceinline__ void cp_async_b128(const float* g, float* l) {
    *reinterpret_cast<float4*>(l) = *reinterpret_cast<const float4*>(g);
}
__device__ __forceinline__ void cp_async_b32(const float* g, float* l) { *l = *g; }
__device__ __forceinline__ void wait_async0() {}
#endif

// ---------------------------------------------------------------- bicubic ---
__device__ __forceinline__ float cubic_w(float t) {
    float at = fabsf(t);
    float w1 = ((A_CUBIC + 2.0f) * at - (A_CUBIC + 3.0f)) * at * at + 1.0f;
    float w2 = A_CUBIC * (((at - 5.0f) * at + 8.0f) * at - 4.0f);
    return at <= 1.0f ? w1 : (at < 2.0f ? w2 : 0.0f);
}

// bicubic, align_corners=True, border clamp; separable 4x4 done directly
__global__ void resize_bicubic(const float* __restrict__ in, float* __restrict__ out,
                               int ohw) {
    int total = BATCH * CH * ohw * ohw;
    int gid = blockIdx.x * blockDim.x + threadIdx.x;
    if (gid >= total) return;
    int ox = gid % ohw;
    int oy = (gid / ohw) % ohw;
    int pl = gid / (ohw * ohw);
    float scale = (float)(H - 1) / (float)(ohw - 1);

    float cy = (float)oy * scale;
    int   iy0 = (int)floorf(cy);
    float fy = cy - floorf(cy);
    float cx = (float)ox * scale;
    int   ix0 = (int)floorf(cx);
    float fx = cx - floorf(cx);

    float wy[4], wx[4];
    int   ry[4], rx[4];
#pragma unroll
    for (int k = 0; k < 4; ++k) {
        int off = k - 1;
        wy[k] = cubic_w(fy - (float)off);
        wx[k] = cubic_w(fx - (float)off);
        int yy = iy0 + off; ry[k] = yy < 0 ? 0 : (yy > H - 1 ? H - 1 : yy);
        int xx = ix0 + off; rx[k] = xx < 0 ? 0 : (xx > W - 1 ? W - 1 : xx);
    }
    const float* base = in + (size_t)pl * H * W;
    float acc = 0.0f;
#pragma unroll
    for (int a = 0; a < 4; ++a) {
        float rowacc = 0.0f;
#pragma unroll
        for (int bq = 0; bq < 4; ++bq)
            rowacc += wx[bq] * base[ry[a] * W + rx[bq]];
        acc += wy[a] * rowacc;
    }
    out[(size_t)pl * ohw * ohw + (size_t)oy * ohw + ox] = acc;
}

// --------------------------------------------------- sfeat = tpf + ipf ------
__global__ void build_sfeat(const float* __restrict__ inp, const float* __restrict__ tar,
                            float* __restrict__ sfeat) {
    int gid = blockIdx.x * blockDim.x + threadIdx.x;   // over BATCH*NP (exact)
    int b = gid / NP, n = gid % NP;
    int top = 3 * (n / NPW), left = 3 * (n % NPW);
    float* row = sfeat + (size_t)gid * D_PAD;
#pragma unroll
    for (int c = 0; c < CH; ++c)
#pragma unroll
        for (int pi = 0; pi < PS; ++pi)
#pragma unroll
            for (int pj = 0; pj < PS; ++pj) {
                int src = ((b * CH + c) * H + top + pi) * W + left + pj;
                row[c * 9 + pi * 3 + pj] = inp[src] + tar[src];
            }
#pragma unroll
    for (int d = 27; d < D_PAD; ++d) row[d] = 0.0f;
}

// -------------------------------------------------------- database build ----
__global__ void build_gf(const float* __restrict__ tar, const float* __restrict__ tar2,
                         const float* __restrict__ tar4,
                         float* __restrict__ gf, float* __restrict__ gn) {
    int gid = blockIdx.x * blockDim.x + threadIdx.x;   // over BATCH*G_PAD (exact)
    int b = gid / G_PAD, g = gid % G_PAD;
    float* grow = gf + (size_t)gid * D_PAD;
    if (g >= G_REAL) {
#pragma unroll
        for (int d = 0; d < D_PAD; ++d) grow[d] = 0.0f;
        gn[gid] = __builtin_inff();
        return;
    }
    // 15 segments: 4 shifts x 3 scales (shifted crops), then 3 full unfolds
    const int seg_off[16] = {0, 2209, 2738, 2859, 5068, 5597, 5718, 7927, 8456,
                             8577, 10786, 11315, 11436, 13740, 14316, 14460};
    int k = 0;
#pragma unroll
    for (int t = 1; t < 15; ++t) if (g >= seg_off[t]) k = t;
    int local = g - seg_off[k];
    int sc, ii, jj, gw;
    if (k < 12) {
        int si = k / 3; sc = k % 3;
        ii = 1 + si / 2; jj = 1 + (si & 1);
        gw = (sc == 0) ? 47 : (sc == 1) ? 23 : 11;
    } else {
        sc = k - 12; ii = 0; jj = 0;
        gw = (sc == 0) ? 48 : (sc == 1) ? 24 : 12;
    }
    const float* img = (sc == 0) ? tar : (sc == 1) ? tar2 : tar4;
    int iw = (sc == 0) ? 144 : (sc == 1) ? 72 : 36;
    int p = local / gw, q = local % gw;
    int top = ii + 3 * p, left = jj + 3 * q;
    float s2 = 0.0f;
#pragma unroll
    for (int c = 0; c < CH; ++c)
#pragma unroll
        for (int pi = 0; pi < PS; ++pi)
#pragma unroll
            for (int pj = 0; pj < PS; ++pj) {
                float v = img[((b * CH + c) * iw + top + pi) * iw + left + pj];
                grow[c * 9 + pi * 3 + pj] = v;
                s2 += v * v;
            }
#pragma unroll
    for (int d = 27; d < D_PAD; ++d) grow[d] = 0.0f;
    gn[gid] = s2;
}

// ---------------- fused WMMA f32 GEMM + argmin, async-LDS double buffering --
// score(n,g) = gn[g] - dot(sfeat[n], gf[g])   (alpha=0.5 constants dropped)
__global__ __launch_bounds__(256) void gemm_argmin(
    const float* __restrict__ sfeat,   // [B][NP][32]
    const float* __restrict__ gf,      // [B][G_PAD][32]
    const float* __restrict__ gn,      // [B][G_PAD]
    int* __restrict__ amin) {          // [B][NP]
    __shared__ __align__(16) float gfbuf[2][CCH * LDS_ROW];
    __shared__ __align__(16) float gnbuf[2][CCH];

    const int tid  = threadIdx.x;
    const int lane = tid & 31;
    const int wave = tid >> 5;
    const int b    = blockIdx.x / BLOCKS_PER_BATCH;   // whole block in one batch
    const int tile = blockIdx.x * 8 + wave;
    const int nbase = (tile % TILES_PER_BATCH) * 16;
    const int l = lane & 15, hh = lane >> 4;

    const float* gfb = gf + (size_t)b * G_PAD * D_PAD;
    const float* gnb = gn + (size_t)b * G_PAD;

    // A fragment (16x4 f32 per step): lane holds row M=l, K = 2*hh + {0,1}
    v2f a[8];
    const float* srow = sfeat + (size_t)(b * NP + nbase + l) * D_PAD;
#pragma unroll
    for (int t = 0; t < 8; ++t) {
        a[t].x = srow[4 * t + 2 * hh];
        a[t].y = srow[4 * t + 2 * hh + 1];
    }

    float best[8]; int bidx[8];
#pragma unroll
    for (int r = 0; r < 8; ++r) { best[r] = __builtin_inff(); bidx[r] = 0; }

    // cooperative stage of one 128-column chunk: 128 rows x 128B = 16KB (+gn)
    auto stage = [&](int bufi, int cc) {
        const float* gsrc = gfb + (size_t)cc * CCH * D_PAD;
#pragma unroll
        for (int k = 0; k < 4; ++k) {
            int p = tid + 256 * k;       // 16B piece id, 1024 pieces
            int r = p >> 3, s = p & 7;
            cp_async_b128(gsrc + r * D_PAD + s * 4,
                          &gfbuf[bufi][r * LDS_ROW + s * 4]);
        }
        if (tid < CCH)
            cp_async_b32(gnb + (size_t)cc * CCH + tid, &gnbuf[bufi][tid]);
    };

    stage(0, 0);
    for (int cc = 0; cc < NCC; ++cc) {
        wait_async0();
        __syncthreads();                 // chunk cc resident; prev reads done
        if (cc + 1 < NCC) stage((cc + 1) & 1, cc + 1);   // overlap DMA
        const float* buf = gfbuf[cc & 1];
        const float* gvn = gnbuf[cc & 1];
#pragma unroll 2
        for (int sub = 0; sub < CCH / 16; ++sub) {
            const float* brow = buf + (sub * 16 + l) * LDS_ROW;
            v2f bf[8];
#pragma unroll
            for (int t = 0; t < 8; ++t) {
                bf[t].x = brow[4 * t + 2 * hh];
                bf[t].y = brow[4 * t + 2 * hh + 1];
            }
            float gnv = gvn[sub * 16 + l];
            v8f acc = {0.f, 0.f, 0.f, 0.f, 0.f, 0.f, 0.f, 0.f};
#pragma unroll
            for (int t = 0; t < 8; ++t)
                acc = __builtin_amdgcn_wmma_f32_16x16x4_f32(
                    false, a[t], false, bf[t], (short)0, acc, false, false);
            int gcol = cc * CCH + sub * 16 + l;
            // C layout: VGPR r -> row (r + 8*hh), column = l
#pragma unroll
            for (int r = 0; r < 8; ++r) {
                float s = gnv - acc[r];
                if (s < best[r]) { best[r] = s; bidx[r] = gcol; }
            }
        }
    }
    // reduce over the 16 lanes of each half (columns), lowest-index tiebreak
#pragma unroll
    for (int mask = 1; mask <= 8; mask <<= 1) {
#pragma unroll
        for (int r = 0; r < 8; ++r) {
            float ov = __shfl_xor(best[r], mask, 32);
            int   oi = __shfl_xor(bidx[r], mask, 32);
            if (ov < best[r] || (ov == best[r] && oi < bidx[r])) {
                best[r] = ov; bidx[r] = oi;
            }
        }
    }
    if (l == 0) {
#pragma unroll
        for (int r = 0; r < 8; ++r)
            amin[b * NP + nbase + r + 8 * hh] = bidx[r];
    }
}

// ------------------------------------------- gather + fold + L1 partials ---
__global__ void select_loss(const float* __restrict__ inp, const float* __restrict__ gf,
                            const int* __restrict__ amin,
                            float* __restrict__ sel_img,   // d_out + 1
                            float* __restrict__ partial) {
    int gid = blockIdx.x * blockDim.x + threadIdx.x;   // over BATCH*NP (exact)
    int b = gid / NP, n = gid % NP;
    int idx = amin[gid];
    const float* grow = gf + (size_t)(b * G_PAD + idx) * D_PAD;
    int top = 3 * (n / NPW), left = 3 * (n % NPW);
    float acc = 0.0f;
#pragma unroll
    for (int c = 0; c < CH; ++c)
#pragma unroll
        for (int pi = 0; pi < PS; ++pi)
#pragma unroll
            for (int pj = 0; pj < PS; ++pj) {
                float sv = grow[c * 9 + pi * 3 + pj];
                int dst = ((b * CH + c) * H + top + pi) * W + left + pj;
                sel_img[dst] = sv;
                acc += fabsf(inp[dst] - sv);
            }
    partial[gid] = acc;
}

__global__ void final_reduce(const float* __restrict__ partial, float* __restrict__ out0) {
    __shared__ float sm[256];
    float s = 0.0f;
    for (int i = threadIdx.x; i < BATCH * NP; i += 256) s += partial[i];
    sm[threadIdx.x] = s;
    __syncthreads();
    for (int off = 128; off > 0; off >>= 1) {
        if ((int)threadIdx.x < off) sm[threadIdx.x] += sm[threadIdx.x + off];
        __syncthreads();
    }
    if (threadIdx.x == 0) out0[0] = sm[0] / (float)(BATCH * CH * H * W);
}

// --------------------------------------------------------------- launcher --
static size_t align256(size_t x) { return (x + 255) & ~(size_t)255; }

extern "C" void kernel_launch(void* const* d_in, const int* in_sizes, int n_in,
                              void* d_out, int out_size, void* d_ws, size_t ws_size,
                              hipStream_t stream) {
    (void)in_sizes; (void)n_in; (void)out_size; (void)ws_size;
    const float* inp = (const float*)d_in[0];
    const float* tar = (const float*)d_in[1];
    float* out = (float*)d_out;

    char* ws = (char*)d_ws;
    size_t off = 0;
    float* tar2    = (float*)(ws + off); off = align256(off + sizeof(float) * BATCH * CH * 72 * 72);
    float* tar4    = (float*)(ws + off); off = align256(off + sizeof(float) * BATCH * CH * 36 * 36);
    float* gfx     = (float*)(ws + off); off = align256(off + sizeof(float) * BATCH * G_PAD * D_PAD);
    float* gnx     = (float*)(ws + off); off = align256(off + sizeof(float) * BATCH * G_PAD);
    float* sfeat   = (float*)(ws + off); off = align256(off + sizeof(float) * BATCH * NP * D_PAD);
    int*   amin    = (int*)(ws + off);   off = align256(off + sizeof(int) * BATCH * NP);
    float* partial = (float*)(ws + off); off = align256(off + sizeof(float) * BATCH * NP);

    // 1) bicubic pyramids
    resize_bicubic<<<(BATCH * CH * 72 * 72 + 255) / 256, 256, 0, stream>>>(tar, tar2, 72);
    resize_bicubic<<<(BATCH * CH * 36 * 36 + 255) / 256, 256, 0, stream>>>(tar, tar4, 36);
    // 2) query features (tpf + ipf), D padded to 32
    build_sfeat<<<(BATCH * NP) / 256, 256, 0, stream>>>(inp, tar, sfeat);
    // 3) database features + squared norms (padded rows -> +inf)
    build_gf<<<(BATCH * G_PAD) / 256, 256, 0, stream>>>(tar, tar2, tar4, gfx, gnx);
    // 4) fused WMMA GEMM + argmin: 288 row-tiles, 8 waves/block, LDS-staged gf
    gemm_argmin<<<(BATCH * TILES_PER_BATCH) / 8, 256, 0, stream>>>(sfeat, gfx, gnx, amin);
    // 5) gather selected patches -> sel_img, per-patch L1 partials
    select_loss<<<(BATCH * NP) / 256, 256, 0, stream>>>(inp, gfx, amin, out + 1, partial);
    // 6) deterministic mean
    final_reduce<<<1, 256, 0, stream>>>(partial, out);
}